// StabilizedShared2ModeModel_80075370267059
// MI455X (gfx1250) — compile-verified
//
#include <hip/hip_runtime.h>
#include <math.h>

typedef __attribute__((ext_vector_type(2))) float v2f;
typedef __attribute__((ext_vector_type(8))) float v8f;

#define B_SZ 256
#define N_SZ 256
#define DM   1024
#define NH   1280   /* n*5 */
#define PI_OVER_24 0.13089969389957471827f

// Fast sigmoid / tanh helpers using single-instruction TRANS ops.
static __device__ __forceinline__ float fast_sigmoid(float x) {
    // 1/(1+e^{-x}) with v_exp_f32 + v_rcp_f32
    return __builtin_amdgcn_rcpf(1.0f + __expf(-x));
}
static __device__ __forceinline__ float fast_tanh_pos(float x) {
#if __has_builtin(__builtin_amdgcn_tanhf)
    return __builtin_amdgcn_tanhf(x);       // v_tanh_f32 (CDNA5 TRANS op)
#else
    // valid for x >= 0 (here x in [0, ~0.4]): tanh(x) = 1 - 2/(e^{2x}+1)
    return 1.0f - 2.0f * __builtin_amdgcn_rcpf(__expf(2.0f * x) + 1.0f);
#endif
}

// -------------------------------------------------------------------------
// Kernel 1: h = history @ W1 + b1 via fp32 WMMA 16x16x4.
// One wave per 16x32 output tile -> TWO independent accumulator chains
// (hides WMMA->WMMA RAW latency) and A-fragment reuse across both chains.
// Grid: (B/16, DM/32) = 512 waves.
// -------------------------------------------------------------------------
__global__ __launch_bounds__(32)
void gemm1_wmma(const float* __restrict__ H, const float* __restrict__ W1,
                const float* __restrict__ b1, float* __restrict__ hbuf) {
    const int row0 = blockIdx.x * 16;
    const int col0 = blockIdx.y * 32;
    const int lane = threadIdx.x;
    const int l    = lane & 15;
    const int half = lane >> 4;

    // A fragment: lane (l,half) holds H[row0+l][k + 2*half + {0,1}]
    const float* aptr = H + (size_t)(row0 + l) * NH + 2 * half;
    // B fragment: lane (l,half) holds W1[k + 2*half + {0,1}][col0 + 16*t + l]
    const float* bptr = W1 + (size_t)(2 * half) * DM + col0 + l;

    v8f acc[2] = {{}, {}};
    for (int k = 0; k < NH; k += 4) {
        if ((k & 127) == 0) {
            __builtin_prefetch(aptr + k + 128, 0, 0);               // global_prefetch_b8
            __builtin_prefetch(bptr + (size_t)(k + 128) * DM, 0, 0);
        }
        v2f a;
        a.x = aptr[k];
        a.y = aptr[k + 1];
        const float* bk0 = bptr + (size_t)k * DM;
        const float* bk1 = bptr + (size_t)(k + 1) * DM;
        v2f b0, b1f;
        b0.x  = bk0[0];   b0.y  = bk1[0];
        b1f.x = bk0[16];  b1f.y = bk1[16];
        acc[0] = __builtin_amdgcn_wmma_f32_16x16x4_f32(false, a, false, b0,
                                                       (short)0, acc[0], false, false);
        acc[1] = __builtin_amdgcn_wmma_f32_16x16x4_f32(false, a, false, b1f,
                                                       (short)0, acc[1], false, false);
    }

#pragma unroll
    for (int t = 0; t < 2; ++t) {
        const int col = col0 + 16 * t + l;
        const float bias = b1[col];
#pragma unroll
        for (int r = 0; r < 8; ++r) {
            hbuf[(size_t)(row0 + r + 8 * half) * DM + col] = acc[t][r] + bias;
        }
    }
}

// -------------------------------------------------------------------------
// Kernel 2: per batch row b: LayerNorm(h) -> ReLU -> @W2 + b2 -> phi/r head
// Grid: B blocks of 256 threads; each thread owns 4 of the 1024 channels.
// -------------------------------------------------------------------------
__global__ __launch_bounds__(256)
void head_kernel(const float* __restrict__ hbuf, const float* __restrict__ ln_g,
                 const float* __restrict__ ln_b, const float* __restrict__ W2,
                 const float* __restrict__ b2,   const float* __restrict__ phi_prev,
                 float* __restrict__ phiR, float* __restrict__ out) {
    __shared__ float red[256];
    __shared__ float red2[256];
    __shared__ float redp[256 * 4];

    const int b = blockIdx.x;
    const int t = threadIdx.x;
    const float* hrow = hbuf + (size_t)b * DM;

    float hv[4];
    hv[0] = hrow[t];
    hv[1] = hrow[t + 256];
    hv[2] = hrow[t + 512];
    hv[3] = hrow[t + 768];

    red[t]  = hv[0] + hv[1] + hv[2] + hv[3];
    red2[t] = hv[0] * hv[0] + hv[1] * hv[1] + hv[2] * hv[2] + hv[3] * hv[3];
    __syncthreads();
    for (int off = 128; off > 0; off >>= 1) {
        if (t < off) { red[t] += red[t + off]; red2[t] += red2[t + off]; }
        __syncthreads();
    }
    const float mu  = red[0] * (1.0f / DM);
    const float var = red2[0] * (1.0f / DM) - mu * mu;
    const float rs  = rsqrtf(var + 1e-5f);

    float p[4] = {0.f, 0.f, 0.f, 0.f};
#pragma unroll
    for (int u = 0; u < 4; ++u) {
        const int idx = t + u * 256;
        float hn = (hv[u] - mu) * rs * ln_g[idx] + ln_b[idx];
        hn = fmaxf(hn, 0.0f);
#pragma unroll
        for (int j = 0; j < 4; ++j) p[j] += hn * W2[idx * 4 + j];
    }
#pragma unroll
    for (int j = 0; j < 4; ++j) redp[t * 4 + j] = p[j];
    __syncthreads();
    for (int off = 128; off > 0; off >>= 1) {
        if (t < off) {
#pragma unroll
            for (int j = 0; j < 4; ++j) redp[t * 4 + j] += redp[(t + off) * 4 + j];
        }
        __syncthreads();
    }

    if (t == 0) {
        // params reshaped (K=2, 2): [k][0] -> d_phi, [k][1] -> r_mode
#pragma unroll
        for (int kk = 0; kk < 2; ++kk) {
            const float p0 = redp[kk * 2 + 0] + b2[kk * 2 + 0];
            const float p1 = redp[kk * 2 + 1] + b2[kk * 2 + 1];
            const float dphi = tanhf(p0) * PI_OVER_24;   // once per row: precision over speed
            const float phi  = phi_prev[b * 2 + kk] + dphi;
            const float rm   = fast_sigmoid(p1);
            phiR[b * 4 + kk]     = phi;   // phi0, phi1
            phiR[b * 4 + 2 + kk] = rm;    // r0,   r1
            out[131072 + b * 2 + kk] = phi;   // phi_curr (after 256*256*2 out0)
            out[131584 + b * 2 + kk] = dphi;  // d_phi
        }
    }
}

// -------------------------------------------------------------------------
// Kernel 3: next[b,i] = sum_j A_gain * rot(theta) * x[b,j]
// Grid: n blocks (one per i), 256 threads (one per b); j-loop of 256.
// Per-i data (softmax weight w0, relu(S)) staged once into LDS, broadcast.
// TRANS-pipe bound: v_cos + v_sin + v_tanh (or v_exp+v_rcp) per cell.
// -------------------------------------------------------------------------
__global__ __launch_bounds__(256)
void mix_kernel(const float* __restrict__ x2, const float* __restrict__ S,
                const float* __restrict__ PW, const float* __restrict__ phiR,
                float* __restrict__ out) {
    __shared__ float w0s[N_SZ];
    __shared__ float srs[N_SZ];

    const int i = blockIdx.x;
    const int b = threadIdx.x;

    {   // softmax over K=2 collapses to a sigmoid: w0 = sigma(p0 - p1)
        const int j = threadIdx.x;
        const float p0 = PW[((size_t)i * N_SZ + j) * 2 + 0];
        const float p1 = PW[((size_t)i * N_SZ + j) * 2 + 1];
        w0s[j] = fast_sigmoid(p0 - p1);
        srs[j] = fmaxf(S[(size_t)i * N_SZ + j], 0.0f);
    }
    __syncthreads();

    const float phi0 = phiR[b * 4 + 0];
    const float phi1 = phiR[b * 4 + 1];
    const float r0   = phiR[b * 4 + 2];
    const float r1   = phiR[b * 4 + 3];
    const float dphi01 = phi0 - phi1;
    const float dr01   = r0 - r1;
    const float* xb  = x2 + (size_t)b * N_SZ * 2;

    float accR = 0.f, accI = 0.f;
#pragma unroll 4
    for (int j = 0; j < N_SZ; ++j) {
        const float w0 = w0s[j];
        const float th = phi1 + w0 * dphi01;       // w0*phi0 + (1-w0)*phi1
        const float rr = r1 + w0 * dr01;           // w0*r0   + (1-w0)*r1
        const float g  = fast_tanh_pos(rr * srs[j]) * 1.1f;  // arg in [0, ~0.4]
        const float cs = __cosf(th);
        const float sn = __sinf(th);
        const float xr = xb[j * 2 + 0];
        const float xi = xb[j * 2 + 1];
        accR += g * (cs * xr - sn * xi);
        accI += g * (sn * xr + cs * xi);
    }
    out[((size_t)b * N_SZ + i) * 2 + 0] = accR;
    out[((size_t)b * N_SZ + i) * 2 + 1] = accI;
}

// -------------------------------------------------------------------------
extern "C" void kernel_launch(void* const* d_in, const int* in_sizes, int n_in,
                              void* d_out, int out_size, void* d_ws, size_t ws_size,
                              hipStream_t stream) {
    const float* x2      = (const float*)d_in[0];   // (256,256,2)
    const float* history = (const float*)d_in[1];   // (256,1280)
    const float* phiprev = (const float*)d_in[2];   // (256,2)
    const float* S       = (const float*)d_in[3];   // (256,256)
    const float* W1      = (const float*)d_in[4];   // (1280,1024)
    const float* b1      = (const float*)d_in[5];   // (1024,)
    const float* lng     = (const float*)d_in[6];   // (1024,)
    const float* lnb     = (const float*)d_in[7];   // (1024,)
    const float* W2      = (const float*)d_in[8];   // (1024,4)
    const float* b2      = (const float*)d_in[9];   // (4,)
    const float* PW      = (const float*)d_in[10];  // (256,256,2)

    float* hbuf = (float*)d_ws;                 // 256*1024 floats = 1 MB
    float* phiR = hbuf + (size_t)B_SZ * DM;     // 256*4 floats
    float* out  = (float*)d_out;                // 131072 + 512 + 512 floats

    gemm1_wmma<<<dim3(B_SZ / 16, DM / 32), 32, 0, stream>>>(history, W1, b1, hbuf);
    head_kernel<<<B_SZ, 256, 0, stream>>>(hbuf, lng, lnb, W2, b2, phiprev, phiR, out);
    mix_kernel<<<N_SZ, 256, 0, stream>>>(x2, S, PW, phiR, out);
}